// ChaoticEmbedding_41824391528937
// MI455X (gfx1250) — compile-verified
//
#include <hip/hip_runtime.h>
#include <math.h>
#include <stdint.h>

// ---------------------------------------------------------------------------
// ChaoticEmbedding for MI455X (gfx1250, wave32)
//
// Roofline: output = 262144*50*3 f32 = 157 MB of stores (~6.7us @ 23.3 TB/s);
// ODE ~0.9G f32 FMA-class ops. Jointly store-BW and f32-VALU bound.
//
// Design:
//  * Layer-1 GEMMs via V_WMMA_F32_16X16X4_F32 (shape-exact: K=4 features),
//    computed transposed (Hidden^T = W^T @ Feat^T) so D columns = particles.
//    2 tiles x 2 packed weight matrices = 4 WMMAs/wave. (verified in asm)
//  * D scattered to LDS (ds_store_b128), each thread gathers its 32 hidden
//    pre-activations; tiny layer-2 heads done scalar.
//  * 49 RK4 Lorenz steps in registers; trajectory staged in a 150 KB LDS
//    tile (256 particles x 150 floats) -> 2 blocks/WGP on CDNA5's 320 KB LDS.
//  * Finalization: each block's tile is one contiguous 153600-byte range in
//    global memory -> single Tensor Data Mover TENSOR_STORE_FROM_LDS (1-D D#)
//    issued by wave 0, waited with s_wait_tensorcnt. Fallback: coalesced
//    float4 copy loop.
// ---------------------------------------------------------------------------

#define WG        256
#define NSTEPS    50
#define HSTEP     0.01f
#define TRAJ_FLOATS (WG * NSTEPS * 3)   // 38400 floats = 153600 bytes LDS

typedef float v2f __attribute__((ext_vector_type(2)));
typedef float v8f __attribute__((ext_vector_type(8)));
typedef unsigned int u32x4 __attribute__((ext_vector_type(4)));
typedef int i32x4 __attribute__((ext_vector_type(4)));
typedef int i32x8 __attribute__((ext_vector_type(8)));

struct S3 { float x, y, z; };

__device__ __forceinline__ float sigmoidf_(float v) {
    return 1.0f / (1.0f + expf(-v));
}

__device__ __forceinline__ S3 lorenz(S3 st, S3 c, float s, float r, float b) {
    S3 d;
    d.x = fmaf(s, st.y - st.x, c.x);                 // s*(y-x) + cx
    d.y = fmaf(st.x, r - st.z, c.y - st.y);          // x*(r-z) - y + cy
    d.z = fmaf(st.x, st.y, fmaf(-b, st.z, c.z));     // x*y - b*z + cz
    return d;
}

__device__ __forceinline__ S3 rk4_step(S3 st, S3 c, float s, float r, float b) {
    S3 k1 = lorenz(st, c, s, r, b);
    S3 t2 = { fmaf(0.5f*HSTEP, k1.x, st.x), fmaf(0.5f*HSTEP, k1.y, st.y), fmaf(0.5f*HSTEP, k1.z, st.z) };
    S3 k2 = lorenz(t2, c, s, r, b);
    S3 t3 = { fmaf(0.5f*HSTEP, k2.x, st.x), fmaf(0.5f*HSTEP, k2.y, st.y), fmaf(0.5f*HSTEP, k2.z, st.z) };
    S3 k3 = lorenz(t3, c, s, r, b);
    S3 t4 = { fmaf(HSTEP, k3.x, st.x), fmaf(HSTEP, k3.y, st.y), fmaf(HSTEP, k3.z, st.z) };
    S3 k4 = lorenz(t4, c, s, r, b);
    const float w = HSTEP / 6.0f;
    S3 o;
    o.x = fmaf(w, k1.x + 2.0f*k2.x + 2.0f*k3.x + k4.x, st.x);
    o.y = fmaf(w, k1.y + 2.0f*k2.y + 2.0f*k3.y + k4.y, st.y);
    o.z = fmaf(w, k1.z + 2.0f*k2.z + 2.0f*k3.z + k4.z, st.z);
    return o;
}

__global__ __launch_bounds__(WG) void chaotic_embed_kernel(
    const float* __restrict__ feat,
    const float* __restrict__ W1,  const float* __restrict__ b1,
    const float* __restrict__ W2,  const float* __restrict__ b2,
    const float* __restrict__ Wc1, const float* __restrict__ bc1,
    const float* __restrict__ Wc2, const float* __restrict__ bc2,
    const float* __restrict__ Wp1, const float* __restrict__ bp1,
    const float* __restrict__ Wp2, const float* __restrict__ bp2,
    float* __restrict__ out)
{
    __shared__ __align__(16) float lds[TRAJ_FLOATS];

    const int tid    = threadIdx.x;
    const int wgBase = blockIdx.x * WG;

    float h[32];   // per-thread layer-1 pre-activations: [0:16)=net1, [16:24)=coup, [24:32)=param

#if __has_builtin(__builtin_amdgcn_wmma_f32_16x16x4_f32)
    {
        const int lane = tid & 31;
        const int wave = tid >> 5;
        const int half = lane >> 4;      // 0: K rows 0,1   1: K rows 2,3
        const int l15  = lane & 15;      // matrix column (particle) / A row (hidden unit)
        const int k0   = 2 * half;

        // A = W1^T  (M=16 hidden units, K=4). W1 stored (4,16) row-major.
        v2f aW1;
        aW1[0] = W1[k0 * 16 + l15];
        aW1[1] = W1[(k0 + 1) * 16 + l15];

        // A = [Wc1^T ; Wp1^T] packed: rows 0..7 coupling hidden, rows 8..15 param hidden.
        v2f aWcp;
        if (l15 < 8) {
            aWcp[0] = Wc1[k0 * 8 + l15];
            aWcp[1] = Wc1[(k0 + 1) * 8 + l15];
        } else {
            aWcp[0] = Wp1[k0 * 8 + (l15 - 8)];
            aWcp[1] = Wp1[(k0 + 1) * 8 + (l15 - 8)];
        }

        const v8f zero = {};

        #pragma unroll
        for (int t = 0; t < 2; ++t) {
            // Column N = particle within tile; thread that owns it:
            const int colT = wave * 32 + t * 16 + l15;
            const size_t p = (size_t)(wgBase + colT);

            // B = Feat^T (K=4, N=16): lane half selects K rows 0,1 / 2,3.
            v2f bF;
            bF[0] = feat[p * 4 + k0];
            bF[1] = feat[p * 4 + k0 + 1];

            v8f dA = __builtin_amdgcn_wmma_f32_16x16x4_f32(
                false, aW1,  false, bF, (short)0, zero, false, false);
            v8f dB = __builtin_amdgcn_wmma_f32_16x16x4_f32(
                false, aWcp, false, bF, (short)0, zero, false, false);

            // D rows: VGPR v -> hidden unit v + 8*half, column owned by thread colT.
            float* dst = &lds[colT * 32 + 8 * half];
            #pragma unroll
            for (int v = 0; v < 8; ++v) dst[v] = dA[v];
            #pragma unroll
            for (int v = 0; v < 8; ++v) dst[16 + v] = dB[v];
        }
        __syncthreads();
        #pragma unroll
        for (int m = 0; m < 32; ++m) h[m] = lds[tid * 32 + m];
    }
#else
    {
        const size_t p = (size_t)(wgBase + tid);
        const float f0 = feat[p*4+0], f1 = feat[p*4+1], f2 = feat[p*4+2], f3 = feat[p*4+3];
        #pragma unroll
        for (int m = 0; m < 16; ++m)
            h[m] = fmaf(f0, W1[m], fmaf(f1, W1[16+m], fmaf(f2, W1[32+m], f3 * W1[48+m])));
        #pragma unroll
        for (int m = 0; m < 8; ++m) {
            h[16+m] = fmaf(f0, Wc1[m], fmaf(f1, Wc1[8+m], fmaf(f2, Wc1[16+m], f3 * Wc1[24+m])));
            h[24+m] = fmaf(f0, Wp1[m], fmaf(f1, Wp1[8+m], fmaf(f2, Wp1[16+m], f3 * Wp1[24+m])));
        }
        __syncthreads();   // keep barrier structure identical
    }
#endif

    // ---- activations + tiny layer-2 heads (scalar, ~100 FMAs/particle) ----
    float h1[16], hc[8], hp[8];
    #pragma unroll
    for (int m = 0; m < 16; ++m) h1[m] = tanhf(h[m] + b1[m]);
    #pragma unroll
    for (int m = 0; m < 8; ++m)  hc[m] = tanhf(h[16+m] + bc1[m]);
    #pragma unroll
    for (int m = 0; m < 8; ++m)  hp[m] = fmaxf(h[24+m] + bp1[m], 0.0f);

    float ip0 = b2[0],  ip1 = b2[1],  ip2 = b2[2];
    #pragma unroll
    for (int m = 0; m < 16; ++m) {
        ip0 = fmaf(h1[m], W2[m*3+0], ip0);
        ip1 = fmaf(h1[m], W2[m*3+1], ip1);
        ip2 = fmaf(h1[m], W2[m*3+2], ip2);
    }
    float cp0 = bc2[0], cp1 = bc2[1], cp2 = bc2[2];
    float pp0 = bp2[0], pp1 = bp2[1], pp2 = bp2[2];
    #pragma unroll
    for (int m = 0; m < 8; ++m) {
        cp0 = fmaf(hc[m], Wc2[m*3+0], cp0);
        cp1 = fmaf(hc[m], Wc2[m*3+1], cp1);
        cp2 = fmaf(hc[m], Wc2[m*3+2], cp2);
        pp0 = fmaf(hp[m], Wp2[m*3+0], pp0);
        pp1 = fmaf(hp[m], Wp2[m*3+1], pp1);
        pp2 = fmaf(hp[m], Wp2[m*3+2], pp2);
    }

    S3 st = { 2.0f * tanhf(ip0), 2.0f * tanhf(ip1), 2.0f * tanhf(ip2) };
    S3 cp = { tanhf(cp0), tanhf(cp1), tanhf(cp2) };   // COUPLING_STRENGTH = 1
    const float sv = 10.0f        * (0.5f + sigmoidf_(pp0));
    const float rv = 28.0f        * (0.5f + sigmoidf_(pp1));
    const float bv = (8.0f/3.0f)  * (0.5f + sigmoidf_(pp2));

    // ---- RK4 integration; stage trajectory in LDS (reuse same buffer) ----
    __syncthreads();   // everyone done reading hidden from LDS before reuse
    float* tr = &lds[tid * (NSTEPS * 3)];
    tr[0] = st.x; tr[1] = st.y; tr[2] = st.z;
    for (int t = 1; t < NSTEPS; ++t) {
        st = rk4_step(st, cp, sv, rv, bv);
        tr[t*3 + 0] = st.x;
        tr[t*3 + 1] = st.y;
        tr[t*3 + 2] = st.z;
    }
    __syncthreads();

    // ---- finalization: block tile is contiguous 153600B in global memory ----
#if defined(__gfx1250__) && __has_builtin(__builtin_amdgcn_tensor_store_from_lds) && __has_builtin(__builtin_amdgcn_s_wait_tensorcnt)
    // Tensor Data Mover: one 1-D descriptor DMAs the whole tile LDS->global.
    if ((threadIdx.x >> 5) == 0) {   // one wave issues; TDM ignores EXEC
        const uint64_t ga = (uint64_t)(uintptr_t)out
                          + (uint64_t)blockIdx.x * (uint64_t)(TRAJ_FLOATS * 4);
        const uint32_t lds_base = (uint32_t)(uintptr_t)&lds[0]; // addr[31:0] = LDS byte offset

        u32x4 g0;
        g0[0] = 1u;                                   // count=1, is_restore=0, gather off
        g0[1] = lds_base;                             // lds_addr (bytes)
        g0[2] = (uint32_t)ga;                         // global_addr[31:0]
        g0[3] = (uint32_t)(ga >> 32) | (2u << 30);    // global_addr[56:32] | type=2 (image)

        i32x8 g1;
        g1[0] = (int)(2u << 16);                      // workgroup_mask=0 (no cluster), data_size=2 (4B)
        g1[1] = (int)((TRAJ_FLOATS & 0xFFFFu) << 16); // tensor_dim0[15:0] in [31:16]
        g1[2] = (int)((TRAJ_FLOATS >> 16) | (1u << 16)); // tensor_dim0[31:16]; tensor_dim1=1
        g1[3] = (int)((TRAJ_FLOATS & 0xFFFFu) << 16); // tensor_dim1[31:16]=0; tile_dim0=38400
        g1[4] = (int)(1u | (1u << 16));               // tile_dim1=1, tile_dim2=1 (single pass)
        g1[5] = (int)TRAJ_FLOATS;                     // tensor_dim0_stride[31:0] (unused, benign)
        g1[6] = 0;                                    // stride0[47:32]=0, stride1[15:0]=0
        g1[7] = 0;                                    // stride1[47:16]=0

        i32x4 g2; g2[0] = 1; g2[1] = 0; g2[2] = 0; g2[3] = 0;  // tensor_dim2=1, no iterate
        i32x4 g3; g3[0] = 0; g3[1] = 0; g3[2] = 0; g3[3] = 0;

 #if __clang_major__ >= 23
        i32x8 g4 = {};
        __builtin_amdgcn_tensor_store_from_lds(g0, g1, g2, g3, g4, 0);
 #else
        __builtin_amdgcn_tensor_store_from_lds(g0, g1, g2, g3, 0);
 #endif
        __builtin_amdgcn_s_wait_tensorcnt((short)0); // keep LDS alive until DMA drains
    }
#else
    // Fallback: fully coalesced float4 bulk copy.
    {
        float4*       o4 = (float4*)(out + (size_t)blockIdx.x * TRAJ_FLOATS);
        const float4* l4 = (const float4*)lds;
        for (int i = tid; i < TRAJ_FLOATS / 4; i += WG)
            o4[i] = l4[i];
    }
#endif
}

extern "C" void kernel_launch(void* const* d_in, const int* in_sizes, int n_in,
                              void* d_out, int out_size, void* d_ws, size_t ws_size,
                              hipStream_t stream) {
    const float* feat = (const float*)d_in[0];
    const float* W1   = (const float*)d_in[1];
    const float* b1   = (const float*)d_in[2];
    const float* W2   = (const float*)d_in[3];
    const float* b2   = (const float*)d_in[4];
    const float* Wc1  = (const float*)d_in[5];
    const float* bc1  = (const float*)d_in[6];
    const float* Wc2  = (const float*)d_in[7];
    const float* bc2  = (const float*)d_in[8];
    const float* Wp1  = (const float*)d_in[9];
    const float* bp1  = (const float*)d_in[10];
    const float* Wp2  = (const float*)d_in[11];
    const float* bp2  = (const float*)d_in[12];
    float* out        = (float*)d_out;

    const int B = in_sizes[0] / 4;        // 262144
    const int nblocks = B / WG;           // exact: 1024

    chaotic_embed_kernel<<<dim3(nblocks), dim3(WG), 0, stream>>>(
        feat, W1, b1, W2, b2, Wc1, bc1, Wc2, bc2, Wp1, bp1, Wp2, bp2, out);
}